// FNO_31610959298930
// MI455X (gfx1250) — compile-verified
//
#include <hip/hip_runtime.h>
#include <hip/hip_bf16.h>
#include <math.h>

// FNO forward for MI455X (gfx1250, wave32).
// Sizes: B=8, N=256 -> BN=2048 rows; T=1024; H=32; H/2=16; MODES=16; L=4.
// Spectral conv realized as basis GEMMs with V_WMMA_F32_16X16X4_F32.
// Tiles staged through LDS with GLOBAL_LOAD_ASYNC_TO_LDS_B128 /
// GLOBAL_STORE_ASYNC_FROM_LDS_B128 (ASYNCcnt) when available.
// Workspace layout (floats):
//   [0      .. 32767 ]  Bf  : forward basis  [1024 t][32 j] (j<16: cos, j>=16: -sin)
//   [32768  .. 65535 ]  Bi  : inverse basis  [32 j][1024 t]
//   [65536  .. +2048*32*1024] x : state [bn][c][t]  (256 MB)

#define BN_TOT 2048
#define TT     1024
#define HH     32
#define MODES  16

typedef __attribute__((ext_vector_type(2))) float v2f;
typedef __attribute__((ext_vector_type(8))) float v8f;
typedef __attribute__((ext_vector_type(4))) int v4i;

#define GASP __attribute__((address_space(1)))
#define LASP __attribute__((address_space(3)))

#if defined(__has_builtin)
#if __has_builtin(__builtin_amdgcn_global_load_async_to_lds_b128) && \
    __has_builtin(__builtin_amdgcn_global_store_async_from_lds_b128)
#define USE_ASYNC_LDS 1
#endif
#endif
#ifndef USE_ASYNC_LDS
#define USE_ASYNC_LDS 0
#endif

__device__ __forceinline__ void wait_asynccnt0() {
#if USE_ASYNC_LDS
#if __has_builtin(__builtin_amdgcn_s_wait_asynccnt)
  __builtin_amdgcn_s_wait_asynccnt(0);
#else
  asm volatile("s_wait_asynccnt 0x0" ::: "memory");
#endif
#endif
}

__device__ __forceinline__ v8f wmma4(v2f a, v2f b, v8f c) {
  // D = A(16x4) * B(4x16) + C(16x16), fp32
  return __builtin_amdgcn_wmma_f32_16x16x4_f32(
      /*neg_a=*/false, a, /*neg_b=*/false, b,
      /*c_mod=*/(short)0, c, /*reuse_a=*/false, /*reuse_b=*/false);
}

__device__ __forceinline__ float gelu_f(float v) {
  // exact erf GELU (torch nn.GELU default)
  return 0.5f * v * (1.0f + erff(v * 0.70710678118654752440f));
}

// ---------------------------------------------------------------- basis tables
__global__ __launch_bounds__(256) void build_basis(float* __restrict__ Bf,
                                                   float* __restrict__ Bi) {
  int idx = blockIdx.x * 256 + threadIdx.x;   // 0..32767
  if (idx >= TT * HH) return;
  const float TWO_PI = 6.283185307179586476925286766559f;
  // forward basis: Bf[t*32 + j]
  {
    int t = idx >> 5, j = idx & 31;
    int m = j & 15;
    float th = TWO_PI * (float)m * (float)t * (1.0f / (float)TT);
    Bf[idx] = (j < MODES) ? cosf(th) : -sinf(th);
  }
  // inverse basis: Bi[j*1024 + t]
  {
    int j = idx >> 10, t = idx & 1023;
    int m = j & 15;
    float th = TWO_PI * (float)m * (float)t * (1.0f / (float)TT);
    float v;
    if (j < MODES) v = (m == 0 ? 1.0f : 2.0f * cosf(th)) * (1.0f / (float)TT);
    else           v = (m == 0 ? 0.0f : -2.0f * sinf(th)) * (1.0f / (float)TT);
    Bi[idx] = v;
  }
}

// ------------------------------------------------- encoder + input 1x1 conv
// x[bn][o][t] = in_w[o][0]*X[bn][t] + cc[bn][o],
// cc[bn][o]   = in_b[o] + sum_h in_w[o][1+h] * (enc(W)[bn][h])
__global__ __launch_bounds__(256) void fno_encode(
    const float* __restrict__ X, const float* __restrict__ W,
    const float* __restrict__ enc_w, const float* __restrict__ enc_b,
    const float* __restrict__ in_w, const float* __restrict__ in_b,
    float* __restrict__ x) {
  __shared__ float part[256];
  __shared__ float wf[16];
  __shared__ float cc[32];
  const int bn = blockIdx.x;
  const int tid = threadIdx.x;

  // Wf[hh] = W[bn,:] . enc_w[hh,:] + enc_b[hh]   (16 dots of length 256)
  {
    int hh = tid >> 4, seg = tid & 15;
    const float* wrow = W + (size_t)bn * 256;
    float p = 0.0f;
    #pragma unroll
    for (int q = 0; q < 16; ++q) {
      int mm = seg * 16 + q;
      p += wrow[mm] * enc_w[hh * 256 + mm];
    }
    part[tid] = p;
  }
  __syncthreads();
  if (tid < 16) {
    float s = enc_b[tid];
    #pragma unroll
    for (int q = 0; q < 16; ++q) s += part[tid * 16 + q];
    wf[tid] = s;
  }
  __syncthreads();
  if (tid < 32) {
    float s = in_b[tid];
    #pragma unroll
    for (int q = 0; q < 16; ++q) s += in_w[tid * 17 + 1 + q] * wf[q];
    cc[tid] = s;
  }
  __syncthreads();

  const float4* x4 = (const float4*)(X + (size_t)bn * TT);
  float4* g4 = (float4*)(x + (size_t)bn * (HH * TT));
  for (int j = 0; j < 32; ++j) {
    int idx4 = tid + j * 256;         // 256 float4 per channel row
    int o = idx4 >> 8, t4 = idx4 & 255;
    float4 v = x4[t4];
    float w0 = in_w[o * 17];
    float c = cc[o];
    float4 r;
    r.x = w0 * v.x + c; r.y = w0 * v.y + c;
    r.z = w0 * v.z + c; r.w = w0 * v.w + c;
    g4[idx4] = r;
  }
}

// --------------------------------------------------------- fused FNO layer
// One workgroup (8 wave32) owns one bn row; full [32 x 1024] tile in LDS.
__global__ __launch_bounds__(256) void fno_layer(
    float* __restrict__ x, const float* __restrict__ Bf,
    const float* __restrict__ Bi, const float* __restrict__ swr,
    const float* __restrict__ swi, const float* __restrict__ cw,
    const float* __restrict__ cb) {
  __shared__ __align__(16) float lx[HH * TT];    // 128 KB state tile [c][t]
  __shared__ float lxf[2][HH][HH];               // FFT partials (two K halves)
  __shared__ float lof[HH][HH];                  // mixed coeffs [o][j]
  __shared__ __align__(16) float lcw[HH * HH];   // conv weights [o][c]
  __shared__ float lcb[HH];

  const int bn = blockIdx.x;
  const int tid = threadIdx.x;
  float* gx = x + (size_t)bn * (HH * TT);

  // ---- stage tile (async DMA into LDS) + weights
#if USE_ASYNC_LDS
  {
    char* gsrc = (char*)gx;
    char* lbase = (char*)lx;
    #pragma unroll
    for (int j = 0; j < 32; ++j) {
      int idx4 = tid + j * 256;
      __builtin_amdgcn_global_load_async_to_lds_b128(
          (GASP v4i*)(gsrc + (size_t)idx4 * 16),
          (LASP v4i*)(lbase + (size_t)idx4 * 16), 0, 0);
    }
  }
#else
  {
    const float4* g4 = (const float4*)gx;
    float4* l4 = (float4*)lx;
    #pragma unroll
    for (int j = 0; j < 32; ++j) l4[tid + j * 256] = g4[tid + j * 256];
  }
#endif
  for (int j = tid; j < HH * HH; j += 256) lcw[j] = cw[j];
  if (tid < HH) lcb[tid] = cb[tid];
  wait_asynccnt0();
  __syncthreads();

  const int lane = tid & 31;
  const int wv = tid >> 5;
  const int m = lane & 15;        // row-in-tile / col-in-tile
  const int h = lane >> 4;        // lane half

  // ---- Phase A: xf[32c][32j] = x[32,1024] . Bf[1024,32]  (WMMA, K split 2 ways)
  {
    const int mi = wv & 1, ni = (wv >> 1) & 1, kh = wv >> 2;
    const int row = mi * 16 + m;
    const int ncol = ni * 16 + m;
    const float* xr = &lx[row * TT];
    v8f acc = {0.f, 0.f, 0.f, 0.f, 0.f, 0.f, 0.f, 0.f};
    #pragma unroll 8
    for (int k0 = kh * 512; k0 < kh * 512 + 512; k0 += 4) {
      v2f a, b;
      a.x = xr[k0 + 2 * h];
      a.y = xr[k0 + 2 * h + 1];
      b.x = Bf[(k0 + 2 * h) * 32 + ncol];
      b.y = Bf[(k0 + 2 * h + 1) * 32 + ncol];
      acc = wmma4(a, b, acc);
    }
    #pragma unroll
    for (int r = 0; r < 8; ++r)
      lxf[kh][mi * 16 + r + 8 * h][ni * 16 + m] = acc[r];
  }
  __syncthreads();

  // ---- Phase B: per-mode complex channel mix (tiny; VALU)
  // of[o][m] (real), of[o][16+m] (imag)
  #pragma unroll
  for (int q = 0; q < 2; ++q) {
    int idx = tid * 2 + q;          // 0..511 = 32 o x 16 modes
    int o = idx >> 4, md = idx & 15;
    float re = 0.f, im = 0.f;
    #pragma unroll 8
    for (int i = 0; i < HH; ++i) {
      float xr = lxf[0][i][md] + lxf[1][i][md];
      float xi = lxf[0][i][16 + md] + lxf[1][i][16 + md];
      float wr = swr[(i * HH + o) * MODES + md];
      float wi = swi[(i * HH + o) * MODES + md];
      re += xr * wr - xi * wi;
      im += xr * wi + xi * wr;
    }
    lof[o][md] = re;
    lof[o][16 + md] = im;
  }
  __syncthreads();

  // ---- Phase C: out[o][t] = gelu( of.Bi + cw.x + cb ) + x
  // Each wave owns whole 16-column t-blocks (both 16-row tiles), so B
  // fragments are reused across two WMMAs and results can be written back
  // into the LDS tile without cross-wave hazards.
  for (int tb = wv; tb < 64; tb += 8) {
    const int t0 = tb * 16;
    v8f acc0 = {0.f, 0.f, 0.f, 0.f, 0.f, 0.f, 0.f, 0.f};
    v8f acc1 = {0.f, 0.f, 0.f, 0.f, 0.f, 0.f, 0.f, 0.f};
    const float* ar0 = &lof[m][0];
    const float* ar1 = &lof[16 + m][0];
    #pragma unroll
    for (int k = 0; k < 32; k += 4) {             // inverse-basis GEMM
      v2f b, a0, a1;
      b.x = Bi[(k + 2 * h) * TT + t0 + m];
      b.y = Bi[(k + 2 * h + 1) * TT + t0 + m];
      a0.x = ar0[k + 2 * h]; a0.y = ar0[k + 2 * h + 1];
      a1.x = ar1[k + 2 * h]; a1.y = ar1[k + 2 * h + 1];
      acc0 = wmma4(a0, b, acc0);
      acc1 = wmma4(a1, b, acc1);
    }
    const float* cr0 = &lcw[m * HH];
    const float* cr1 = &lcw[(16 + m) * HH];
    #pragma unroll
    for (int k = 0; k < 32; k += 4) {             // pointwise-conv GEMM
      v2f b, a0, a1;
      b.x = lx[(k + 2 * h) * TT + t0 + m];
      b.y = lx[(k + 2 * h + 1) * TT + t0 + m];
      a0.x = cr0[k + 2 * h]; a0.y = cr0[k + 2 * h + 1];
      a1.x = cr1[k + 2 * h]; a1.y = cr1[k + 2 * h + 1];
      acc0 = wmma4(a0, b, acc0);
      acc1 = wmma4(a1, b, acc1);
    }
    // epilogue: bias + GELU + residual, written back into the LDS tile
    #pragma unroll
    for (int r = 0; r < 8; ++r) {
      int t = t0 + m;
      int o0 = r + 8 * h;
      int o1 = 16 + r + 8 * h;
      float s0 = acc0[r] + lcb[o0];
      float s1 = acc1[r] + lcb[o1];
      float n0 = gelu_f(s0) + lx[o0 * TT + t];
      float n1 = gelu_f(s1) + lx[o1 * TT + t];
      lx[o0 * TT + t] = n0;
      lx[o1 * TT + t] = n1;
    }
  }
  __syncthreads();

  // ---- drain tile back to HBM (coalesced 16B/lane async DMA stores)
#if USE_ASYNC_LDS
  {
    char* gdst = (char*)gx;
    char* lbase = (char*)lx;
    #pragma unroll
    for (int j = 0; j < 32; ++j) {
      int idx4 = tid + j * 256;
      __builtin_amdgcn_global_store_async_from_lds_b128(
          (GASP v4i*)(gdst + (size_t)idx4 * 16),
          (LASP v4i*)(lbase + (size_t)idx4 * 16), 0, 0);
    }
  }
  // S_ENDPGM performs an implicit wait-idle; next dispatch sees the data.
#else
  {
    float4* g4 = (float4*)gx;
    const float4* l4 = (const float4*)lx;
    #pragma unroll
    for (int j = 0; j < 32; ++j) g4[tid + j * 256] = l4[tid + j * 256];
  }
#endif
}

// ------------------------------------------------------- output projection
__global__ __launch_bounds__(256) void fno_out(
    const float* __restrict__ x, const float* __restrict__ w1,
    const float* __restrict__ b1, const float* __restrict__ w2,
    const float* __restrict__ b2, float* __restrict__ out) {
  __shared__ __align__(16) float lx[HH * TT];
  __shared__ float lw1[16 * 32];
  __shared__ float lb1[16];
  __shared__ float lw2[16];
  const int bn = blockIdx.x;
  const int tid = threadIdx.x;
#if USE_ASYNC_LDS
  {
    char* gsrc = (char*)(x + (size_t)bn * (HH * TT));
    char* lbase = (char*)lx;
    #pragma unroll
    for (int j = 0; j < 32; ++j) {
      int idx4 = tid + j * 256;
      __builtin_amdgcn_global_load_async_to_lds_b128(
          (GASP v4i*)(gsrc + (size_t)idx4 * 16),
          (LASP v4i*)(lbase + (size_t)idx4 * 16), 0, 0);
    }
  }
#else
  {
    const float4* g4 = (const float4*)(x + (size_t)bn * (HH * TT));
    float4* l4 = (float4*)lx;
    #pragma unroll
    for (int j = 0; j < 32; ++j) l4[tid + j * 256] = g4[tid + j * 256];
  }
#endif
  for (int j = tid; j < 512; j += 256) lw1[j] = w1[j];
  if (tid < 16) { lb1[tid] = b1[tid]; lw2[tid] = w2[tid]; }
  const float ob2 = b2[0];
  wait_asynccnt0();
  __syncthreads();
  for (int j = 0; j < 4; ++j) {
    int t = tid + j * 256;
    float acc = ob2;
    #pragma unroll
    for (int hh = 0; hh < 16; ++hh) {
      float a = lb1[hh];
      #pragma unroll 8
      for (int c = 0; c < 32; ++c) a += lw1[hh * 32 + c] * lx[c * TT + t];
      acc += lw2[hh] * gelu_f(a);
    }
    out[(size_t)bn * TT + t] = acc;
  }
}

extern "C" void kernel_launch(void* const* d_in, const int* in_sizes, int n_in,
                              void* d_out, int out_size, void* d_ws,
                              size_t ws_size, hipStream_t stream) {
  (void)in_sizes; (void)n_in; (void)out_size; (void)ws_size;
  const float* X      = (const float*)d_in[0];   // [8,256,1024]
  const float* W      = (const float*)d_in[1];   // [8,256,256]
  const float* enc_w  = (const float*)d_in[2];   // [16,256]
  const float* enc_b  = (const float*)d_in[3];   // [16]
  const float* in_w   = (const float*)d_in[4];   // [32,17]
  const float* in_b   = (const float*)d_in[5];   // [32]
  const float* sw_r   = (const float*)d_in[6];   // [4,32,32,16]
  const float* sw_i   = (const float*)d_in[7];   // [4,32,32,16]
  const float* cw     = (const float*)d_in[8];   // [4,32,32]
  const float* cb     = (const float*)d_in[9];   // [4,32]
  const float* out_w1 = (const float*)d_in[10];  // [16,32]
  const float* out_b1 = (const float*)d_in[11];  // [16]
  const float* out_w2 = (const float*)d_in[12];  // [1,16]
  const float* out_b2 = (const float*)d_in[13];  // [1]
  float* out = (float*)d_out;                    // [8,256,1024]

  float* ws = (float*)d_ws;
  float* Bf = ws;                  // 32768 floats
  float* Bi = ws + 32768;          // 32768 floats
  float* x  = ws + 65536;          // 2048*32*1024 floats

  build_basis<<<128, 256, 0, stream>>>(Bf, Bi);
  fno_encode<<<BN_TOT, 256, 0, stream>>>(X, W, enc_w, enc_b, in_w, in_b, x);
  for (int l = 0; l < 4; ++l) {
    fno_layer<<<BN_TOT, 256, 0, stream>>>(
        x, Bf, Bi,
        sw_r + (size_t)l * HH * HH * MODES,
        sw_i + (size_t)l * HH * HH * MODES,
        cw + (size_t)l * HH * HH,
        cb + (size_t)l * HH);
  }
  fno_out<<<BN_TOT, 256, 0, stream>>>(x, out_w1, out_b1, out_w2, out_b2, out);
}